// AdaptiveTokenSampling_2147483648519
// MI455X (gfx1250) — compile-verified
//
#include <hip/hip_runtime.h>
#include <hip/hip_bf16.h>
#include <math.h>

typedef __attribute__((ext_vector_type(2))) float v2f;
typedef __attribute__((ext_vector_type(8))) float v8f;

#define B  8
#define H  12
#define N  1024
#define D  64
#define KS 256
#define MASK_VALUE (-1.7014117e38f)   // -finfo(f32).max / 2

// ---------------------------------------------------------------------------
// Kernel A: cls_score[b][t-1] = sum_h attn[b,h,0,t] * ||value[b,h,t,:]||
// Norms via V_WMMA_F32_16X16X4_F32 Gram-tile diagonal (16 tokens / wave-tile).
// ---------------------------------------------------------------------------
__global__ void ats_norm_score_kernel(const float* __restrict__ attn,
                                      const float* __restrict__ value,
                                      float* __restrict__ cls_score) {
  const int lane = threadIdx.x & 31;
  const int wv   = threadIdx.x >> 5;
  const int tile = blockIdx.x * 8 + wv;     // 0..63, 16 tokens each
  const int b    = blockIdx.y;
  const int tok0 = 1 + tile * 16;           // tokens 1..1023 (tile 63 partial)

  __shared__ float lds[8][16 * 68];         // padded rows: conflict-free
  float* T = lds[wv];

  const int m  = lane & 15;                 // row within tile (A-matrix M)
  const int kh = (lane >> 4) * 2;           // K sub-offset per A layout
  const int myTok = tok0 + lane;            // valid for lanes 0..15
  float score = 0.f;

  for (int h = 0; h < H; ++h) {
    const float* tileBase = value + (((size_t)(b * H + h)) * N + tok0) * D;
    // Stage 16 tokens x 64 floats, fully-coalesced float4 loads (branchless).
#pragma unroll
    for (int i = 0; i < 8; ++i) {
      int f = lane * 4 + i * 128;           // flat float index in 16x64 tile
      int r = f >> 6, cc = f & 63;
      bool ok = (tok0 + r) < N;
      float sel = ok ? 1.f : 0.f;
      float4 v = *(const float4*)(tileBase + (ok ? f : 0));  // always in-bounds
      v.x *= sel; v.y *= sel; v.z *= sel; v.w *= sel;
      *(float4*)&T[r * 68 + cc] = v;
    }
    __syncthreads();

    // Gram tile: C += A * A^T over K=64 in 16 steps of K=4 (f32 WMMA).
    v8f c = {};
#pragma unroll
    for (int ks = 0; ks < 16; ++ks) {
      v2f a;
      a.x = T[m * 68 + ks * 4 + kh];
      a.y = T[m * 68 + ks * 4 + kh + 1];
      c = __builtin_amdgcn_wmma_f32_16x16x4_f32(false, a, false, a,
                                                (short)0, c, false, false);
    }

    // Diagonal (m,m): m<8 -> lane m, vgpr m ; m>=8 -> lane 24+(m-8), vgpr m-8
    // Branchless dynamic extract: masked-FMA sum over the 8 accumulator regs.
    int r = ((lane < 8) ? lane : (lane - 24)) & 7;
    float dv = 0.f;
#pragma unroll
    for (int i = 0; i < 8; ++i)
      dv = fmaf((r == i) ? 1.f : 0.f, c[i], dv);

    int src = (m < 8) ? m : (m + 16);
    float diag = __shfl(dv, src, 32);

    if (lane < 16 && myTok < N) {
      const float* clsRow = attn + ((size_t)(b * H + h)) * N * N;  // row 0
      score += clsRow[myTok] * sqrtf(fmaxf(diag, 0.f));
    }
    __syncthreads();   // before next h overwrites LDS
  }

  if (lane < 16 && myTok < N)
    cls_score[b * N + (myTok - 1)] = score;
}

// ---------------------------------------------------------------------------
// Kernel B: per-batch normalize + log + mask -> logits
// ---------------------------------------------------------------------------
__global__ void ats_logits_kernel(const float* __restrict__ cls_score,
                                  const unsigned char* __restrict__ mask,
                                  float* __restrict__ logits) {
  const int b = blockIdx.x, tid = threadIdx.x;
  __shared__ float red[256];
  float part = 0.f;
  for (int t = tid; t < N - 1; t += 256) part += cls_score[b * N + t];
  red[tid] = part;
  __syncthreads();
  for (int s = 128; s > 0; s >>= 1) {
    if (tid < s) red[tid] += red[tid + s];
    __syncthreads();
  }
  const float S = red[0];
  for (int t = tid; t < N - 1; t += 256) {
    float normed = cls_score[b * N + t] / (S + 1e-6f);
    float lg = logf(normed + 1e-6f);
    bool mk = mask[b * N + t + 1] != 0;
    logits[b * N + t] = mk ? lg : MASK_VALUE;
  }
}

// ---------------------------------------------------------------------------
// Kernel C: gumbel argmax, one wave32 per (b,k); first-index tie-break
// ---------------------------------------------------------------------------
__global__ void ats_argmax_kernel(const float* __restrict__ logits,
                                  const float* __restrict__ gumbel,
                                  int* __restrict__ sampled) {
  const int lane = threadIdx.x & 31;
  const int wv   = threadIdx.x >> 5;
  const int w = blockIdx.x * 8 + wv;        // 0..2047
  const int b = w >> 8, k = w & 255;
  const float* lg = logits + b * N;
  const float* g  = gumbel + ((size_t)(b * KS + k)) * (N - 1);

  float best = -3.402823466e38f;
  int bi = 0x7fffffff;
  for (int t = lane; t < N - 1; t += 32) {
    float v = lg[t] + g[t];
    if (v > best || (v == best && t < bi)) { best = v; bi = t; }
  }
  for (int off = 16; off > 0; off >>= 1) {
    float ob = __shfl_xor(best, off, 32);
    int   oi = __shfl_xor(bi,   off, 32);
    if (ob > best || (ob == best && oi < bi)) { best = ob; bi = oi; }
  }
  if (lane == 0) sampled[b * KS + k] = bi + 1;
}

// ---------------------------------------------------------------------------
// Kernel D: per-batch sort(256) + dedup + compact -> uniq / new_mask
// ---------------------------------------------------------------------------
__global__ void ats_uniq_kernel(const int* __restrict__ sampled,
                                int* __restrict__ uniq_i,
                                float* __restrict__ out_mask,
                                float* __restrict__ out_uniq) {
  const int b = blockIdx.x, tid = threadIdx.x;
  __shared__ int s[256], ps[256], comp[256];
  s[tid] = sampled[b * KS + tid];
  __syncthreads();

  // bitonic sort ascending
  for (int k = 2; k <= 256; k <<= 1) {
    for (int j = k >> 1; j > 0; j >>= 1) {
      int ixj = tid ^ j;
      if (ixj > tid) {
        int a = s[tid], c = s[ixj];
        bool asc = (tid & k) == 0;
        if (a != c && ((a > c) == asc)) { s[tid] = c; s[ixj] = a; }
      }
      __syncthreads();
    }
  }

  int keep = (tid == 0) || (s[tid] != s[tid - 1]);
  ps[tid] = keep;
  __syncthreads();
  for (int off = 1; off < 256; off <<= 1) {   // inclusive scan
    int v = ps[tid] + ((tid >= off) ? ps[tid - off] : 0);
    __syncthreads();
    ps[tid] = v;
    __syncthreads();
  }
  comp[tid] = 0;
  __syncthreads();
  if (keep) comp[ps[tid] - 1] = s[tid];       // compact, ascending, zero-tail
  __syncthreads();

  int u = comp[tid];
  uniq_i[b * (KS + 1) + 1 + tid]   = u;
  out_uniq[b * (KS + 1) + 1 + tid] = (float)u;
  out_mask[b * (KS + 1) + 1 + tid] = (u != 0) ? 1.f : 0.f;
  if (tid == 0) {
    uniq_i[b * (KS + 1)]   = 0;
    out_uniq[b * (KS + 1)] = 0.f;
    out_mask[b * (KS + 1)] = 1.f;
  }
}

// ---------------------------------------------------------------------------
// Kernel E: row gather, one block per (b,h,j) row, float4 copies + prefetch
// ---------------------------------------------------------------------------
__global__ void ats_gather_kernel(const float* __restrict__ attn,
                                  const int* __restrict__ uniq_i,
                                  float* __restrict__ out_attn) {
  const int j = blockIdx.x, h = blockIdx.y, b = blockIdx.z;
  const int tid = threadIdx.x;
  const int row = uniq_i[b * (KS + 1) + j];
  const float4* src =
      (const float4*)(attn + (((size_t)(b * H + h)) * N + row) * N);
  float4* dst =
      (float4*)(out_attn + (((size_t)(b * H + h)) * (KS + 1) + j) * N);
  __builtin_prefetch(src + tid, 0, 0);
  dst[tid] = src[tid];
}

// ---------------------------------------------------------------------------
extern "C" void kernel_launch(void* const* d_in, const int* in_sizes, int n_in,
                              void* d_out, int out_size, void* d_ws, size_t ws_size,
                              hipStream_t stream) {
  const float*         attn   = (const float*)d_in[0];
  const float*         value  = (const float*)d_in[1];
  const unsigned char* mask   = (const unsigned char*)d_in[2];
  const float*         gumbel = (const float*)d_in[3];
  float* out = (float*)d_out;
  float* ws  = (float*)d_ws;

  float* cls_score = ws;                       // 8*1024 floats
  float* logits    = ws + B * N;               // 8*1024 floats
  int*   sampled   = (int*)(ws + 2 * B * N);   // 8*256 ints
  int*   uniq_i    = (int*)(ws + 2 * B * N + B * KS);  // 8*257 ints

  const size_t ATTN_OUT = (size_t)B * H * (KS + 1) * N;   // 25,264,128
  float* out_mask = out + ATTN_OUT;                        // 8*257
  float* out_uniq = out_mask + (size_t)B * (KS + 1);       // 8*257

  ats_norm_score_kernel<<<dim3(8, B), 256, 0, stream>>>(attn, value, cls_score);
  ats_logits_kernel<<<B, 256, 0, stream>>>(cls_score, mask, logits);
  ats_argmax_kernel<<<256, 256, 0, stream>>>(logits, gumbel, sampled);
  ats_uniq_kernel<<<B, 256, 0, stream>>>(sampled, uniq_i, out_mask, out_uniq);
  ats_gather_kernel<<<dim3(KS + 1, H, B), 256, 0, stream>>>(attn, uniq_i, out);
}